// QLSTM_65481071406388
// MI455X (gfx1250) — compile-verified
//
#include <hip/hip_runtime.h>
#include <math.h>

#define T_STEPS 512
#define BATCH   4096
#define DIN     16
#define KDIM    20

typedef __attribute__((ext_vector_type(2))) float v2f;
typedef __attribute__((ext_vector_type(8))) float v8f;

// Flat shared-memory layout (float offsets)
#define U_OFF     0      // 1024 floats: U[g][j][n][2]  (g:4, j:8 subspace cols, n:16, re/im)
#define PRE_OFF   1024   // 512 floats:  pre[slocal(32)][n(16)]   n = g*4 + w
#define ACT_OFF   1536   // 512 floats:  act[slocal(32)][w(4)][g(4)]
#define HX_OFF    2048   // 128 floats:  hx[slocal(32)][w(4)]
#define SMEM_FLOATS 2176

__device__ __forceinline__ float fsigmoid(float x) {
    return 1.f / (1.f + __expf(-x));
}
__device__ __forceinline__ float ftanh(float x) {
    // 1 - 2/(e^{2x}+1): exact saturation via exp inf/0 behavior
    return 1.f - 2.f / (__expf(2.f * x) + 1.f);
}

// ---------------------------------------------------------------------------
// Setup kernel: the variational sub-circuit (RX RY RZ per wire + CNOT chain)
// is a fixed 16x16 unitary U_g per gate.  Encoder input lives in the n2==0
// subspace (Rz on |0> is a global phase), so only 8 columns are needed:
//   j = n0*4 + n1*2 + n3  ->  n = n0*8 + n1*4 + n3.
// One wave; thread (g, j) propagates e_n through the circuit.
// ---------------------------------------------------------------------------
__global__ __launch_bounds__(32)
void qlstm_setup_kernel(const float* __restrict__ q_rx,
                        const float* __restrict__ q_ry,
                        const float* __restrict__ q_rz,
                        float* __restrict__ Uout /* [4][8][16][2] */)
{
    const int tid = threadIdx.x;
    const int g = tid >> 3;
    const int j = tid & 7;
    const int n0 = (j >> 2) & 1, n1 = (j >> 1) & 1, n3 = j & 1;
    const int ninit = n0 * 8 + n1 * 4 + n3;

    float sr[16], si[16];
    #pragma unroll
    for (int n = 0; n < 16; ++n) { sr[n] = 0.f; si[n] = 0.f; }
    sr[ninit] = 1.f;

    #pragma unroll
    for (int w = 0; w < 4; ++w) {
        const int str = 1 << (3 - w);
        float c, s;
        // RX: [[c, -i s],[-i s, c]]
        __sincosf(0.5f * q_rx[g * 4 + w], &s, &c);
        #pragma unroll
        for (int p = 0; p < 8; ++p) {
            const int nlo = ((p & ~(str - 1)) << 1) | (p & (str - 1));
            const int nhi = nlo + str;
            const float lr = sr[nlo], li = si[nlo], hr = sr[nhi], hi = si[nhi];
            sr[nlo] = c * lr + s * hi;  si[nlo] = c * li - s * hr;
            sr[nhi] = c * hr + s * li;  si[nhi] = c * hi - s * lr;
        }
        // RY: [[c, -s],[s, c]]
        __sincosf(0.5f * q_ry[g * 4 + w], &s, &c);
        #pragma unroll
        for (int p = 0; p < 8; ++p) {
            const int nlo = ((p & ~(str - 1)) << 1) | (p & (str - 1));
            const int nhi = nlo + str;
            const float lr = sr[nlo], li = si[nlo], hr = sr[nhi], hi = si[nhi];
            sr[nlo] = c * lr - s * hr;  si[nlo] = c * li - s * hi;
            sr[nhi] = s * lr + c * hr;  si[nhi] = s * li + c * hi;
        }
        // RZ: diag(e^{-i t/2}, e^{+i t/2})
        __sincosf(0.5f * q_rz[g * 4 + w], &s, &c);
        #pragma unroll
        for (int p = 0; p < 8; ++p) {
            const int nlo = ((p & ~(str - 1)) << 1) | (p & (str - 1));
            const int nhi = nlo + str;
            const float lr = sr[nlo], li = si[nlo], hr = sr[nhi], hi = si[nhi];
            sr[nlo] = c * lr + s * li;  si[nlo] = c * li - s * lr;
            sr[nhi] = c * hr - s * hi;  si[nhi] = c * hi + s * hr;
        }
    }
    #pragma unroll
    for (int cwi = 0; cwi < 3; ++cwi) {      // CNOT(w, w+1)
        const int strc = 1 << (3 - cwi);
        const int strt = strc >> 1;
        #pragma unroll
        for (int n = 0; n < 16; ++n) {
            if ((n & strc) && !(n & strt)) {
                const int m = n + strt;
                float tr = sr[n]; sr[n] = sr[m]; sr[m] = tr;
                float ti = si[n]; si[n] = si[m]; si[m] = ti;
            }
        }
    }
    #pragma unroll
    for (int n = 0; n < 16; ++n) {
        Uout[(((g * 8) + j) * 16 + n) * 2 + 0] = sr[n];
        Uout[(((g * 8) + j) * 16 + n) * 2 + 1] = si[n];
    }
}

// ---------------------------------------------------------------------------
// Main persistent kernel: 128 threads (4 waves, wave32) own 32 samples for
// all T steps.
//  - waves 0,1: pre-activation GEMM via V_WMMA_F32_16X16X4_F32 (one 16-sample
//    M-tile per wave, 5 K-chunks, bias as C operand)
//  - all waves: circuit phase, thread = (sample, gate); wave g handles gate g
//    for all 32 samples -> U_g LDS reads are wave-broadcast (conflict-free).
//    Encoder state components are purely real or purely imaginary -> the
//    16x8 complex matvec costs only 2 FMA per complex element (256 FMA).
//  - all threads: LSTM combine, thread = (sample, hidden w); contiguous b32
//    output stores
// ---------------------------------------------------------------------------
__global__ __launch_bounds__(128)
void qlstm_main_kernel(const float* __restrict__ x,     // [T,B,16]
                       const float* __restrict__ Wm,    // [16][20]  n*20+d, n=g*4+q
                       const float* __restrict__ bias,  // [16]
                       const float* __restrict__ Uin,   // [4][8][16][2]
                       float* __restrict__ out)         // outputs | hx | cx
{
    __shared__ __align__(16) float smem[SMEM_FLOATS];

    const int tid  = threadIdx.x;        // 0..127
    const int lane = tid & 31;
    const int wave = tid >> 5;
    const int lo16 = lane & 15;
    const int hi   = lane >> 4;
    const int sBase = blockIdx.x * 32;

    const int g      = wave;             // circuit: this wave's gate
    const int slocal = lane;             // circuit: this lane's sample
    const int cs     = tid >> 2;         // combine: sample
    const int cw     = tid & 3;          // combine: hidden unit  (cs*4+cw == tid)

    // Stage unitaries into LDS (each thread copies 8 floats).
    {
        const float4* up = (const float4*)Uin;
        float4* sp = (float4*)&smem[U_OFF];
        sp[tid * 2 + 0] = up[tid * 2 + 0];
        sp[tid * 2 + 1] = up[tid * 2 + 1];
    }

    // Opaque base offset for U_g: prevents the compiler from proving
    // no-alias vs the in-loop LDS stores and hoisting 4KB of table into VGPRs.
    unsigned ubyte = (unsigned)(U_OFF * 4) + (unsigned)g * 256u;
    asm volatile("" : "+v"(ubyte));
    const float4* Ug = (const float4*)((const char*)smem + ubyte); // [j*8 + k]

    // B operand (W') resident in VGPRs of waves 0,1.
    // ISA 7.12.2 layout: lanes 0-15 carry K rows {0,1} in V0,V1; lanes 16-31 {2,3}.
    v2f Bop[5];
    float biasN = 0.f;
    if (wave < 2) {
        #pragma unroll
        for (int c = 0; c < 5; ++c) {
            const int d = c * 4 + hi * 2;
            Bop[c].x = Wm[lo16 * KDIM + d];
            Bop[c].y = Wm[lo16 * KDIM + d + 1];
        }
        biasN = bias[lo16];
    }

    float cxv = 0.f, hxv = 0.f;          // this thread's (sample, w) cell state
    __syncthreads();

    for (int t = 0; t < T_STEPS; ++t) {
        smem[HX_OFF + tid] = hxv;        // publish hx[cs][cw]
        __syncthreads();                 // B1

        const float* xrow = x + (size_t)t * BATCH * DIN;

        if (wave < 2) {
            // ---- pre = comb x W' + bias via WMMA ----
            const int tile = wave;
            const int m = tile * 16 + lo16;               // A row (local sample)
            const float* xr = xrow + (size_t)(sBase + m) * DIN;
            v8f acc;
            #pragma unroll
            for (int r = 0; r < 8; ++r) acc[r] = biasN;   // C[m][n] = bias[n]
            #pragma unroll
            for (int c = 0; c < 5; ++c) {
                v2f a;
                if (c < 4) {                              // features 0..15 from x_t
                    const int d = c * 4 + hi * 2;
                    a.x = xr[d]; a.y = xr[d + 1];
                } else {                                  // features 16..19 = hx
                    a.x = smem[HX_OFF + m * 4 + hi * 2];
                    a.y = smem[HX_OFF + m * 4 + hi * 2 + 1];
                }
                acc = __builtin_amdgcn_wmma_f32_16x16x4_f32(
                    false, a, false, Bop[c], (short)0, acc, false, false);
            }
            // D: lane owns column n=lo16, rows r + 8*hi -> spill by sample
            #pragma unroll
            for (int r = 0; r < 8; ++r)
                smem[PRE_OFF + (tile * 16 + r + 8 * hi) * 16 + lo16] = acc[r];
        } else if (t + 1 < T_STEPS) {
            // waves 2,3: prefetch next step's x tile (64 threads x 32B = 2KB)
            __builtin_prefetch(xrow + (size_t)BATCH * DIN +
                               (size_t)sBase * DIN + (size_t)(tid - 64) * 8, 0, 1);
        }
        __syncthreads();                 // B2

        // ---- quantum circuit: phi = U_g * psi_encoder ----
        {
            const float4 angv = *(const float4*)&smem[PRE_OFF + slocal * 16 + 4 * g];
            float c0, s0, c1, s1, c3, s3;
            __sincosf(0.5f * angv.x, &s0, &c0);   // RX wire0
            __sincosf(0.5f * angv.y, &s1, &c1);   // RY wire1
            __sincosf(0.5f * angv.w, &s3, &c3);   // RX wire3 (RZ wire2 = global phase)

            // psi_j is purely real (j=0,2,5,7) or purely imaginary (j=1,3,4,6):
            //   a = (c0, -i s0), b = (c1, s1) real, d = (c3, -i s3)
            const float A  = c0 * c3, Bq = c0 * s3;
            const float C  = s0 * c3, D  = s0 * s3;
            const float val[8] = {  A * c1, -Bq * c1,  A * s1, -Bq * s1,
                                   -C * c1, -D * c1,  -C * s1, -D * s1 };
            const bool  im[8]  = { false, true, false, true, true, false, true, false };

            float phr[16], phim[16];
            #pragma unroll
            for (int n = 0; n < 16; ++n) { phr[n] = 0.f; phim[n] = 0.f; }
            #pragma unroll
            for (int jj = 0; jj < 8; ++jj) {
                const float v = val[jj];
                #pragma unroll
                for (int k = 0; k < 8; ++k) {
                    const float4 u = Ug[jj * 8 + k];   // wave-broadcast ds_load_b128
                    const int n = 2 * k;
                    if (!im[jj]) {                     // psi_j = v (real)
                        phr[n]      = fmaf(v, u.x, phr[n]);
                        phim[n]     = fmaf(v, u.y, phim[n]);
                        phr[n + 1]  = fmaf(v, u.z, phr[n + 1]);
                        phim[n + 1] = fmaf(v, u.w, phim[n + 1]);
                    } else {                           // psi_j = i*v
                        phr[n]      = fmaf(-v, u.y, phr[n]);
                        phim[n]     = fmaf( v, u.x, phim[n]);
                        phr[n + 1]  = fmaf(-v, u.w, phr[n + 1]);
                        phim[n + 1] = fmaf( v, u.z, phim[n + 1]);
                    }
                }
            }
            // probabilities + signed-sum butterfly:
            //   z_w = sum_n (-1)^{bit_{3-w}(n)} p[n]
            float p0[16];
            #pragma unroll
            for (int n = 0; n < 16; ++n)
                p0[n] = fmaf(phr[n], phr[n], phim[n] * phim[n]);

            float s1v[8], z3 = 0.f;
            #pragma unroll
            for (int m2 = 0; m2 < 8; ++m2) {
                z3 += p0[2 * m2] - p0[2 * m2 + 1];
                s1v[m2] = p0[2 * m2] + p0[2 * m2 + 1];
            }
            float s2v[4], z2 = 0.f;
            #pragma unroll
            for (int m2 = 0; m2 < 4; ++m2) {
                z2 += s1v[2 * m2] - s1v[2 * m2 + 1];
                s2v[m2] = s1v[2 * m2] + s1v[2 * m2 + 1];
            }
            const float z1 = (s2v[0] - s2v[1]) + (s2v[2] - s2v[3]);
            const float z0 = (s2v[0] + s2v[1]) - (s2v[2] + s2v[3]);

            smem[ACT_OFF + slocal * 16 + 0 * 4 + g] = angv.x + z0;
            smem[ACT_OFF + slocal * 16 + 1 * 4 + g] = angv.y + z1;
            smem[ACT_OFF + slocal * 16 + 2 * 4 + g] = angv.z + z2;
            smem[ACT_OFF + slocal * 16 + 3 * 4 + g] = angv.w + z3;
        }
        __syncthreads();                 // B3

        // ---- LSTM combine: thread = (cs, cw) ----
        {
            const float4 a4 = *(const float4*)&smem[ACT_OFF + cs * 16 + cw * 4];
            const float f  = fsigmoid(a4.x);
            const float i_ = fsigmoid(a4.y);
            const float gg = ftanh(a4.z);
            const float o  = fsigmoid(a4.w);
            cxv = f * cxv + i_ * gg;
            hxv = o * ftanh(cxv);
            // (t*B + sBase + cs)*4 + cw  ==  t*B*4 + sBase*4 + tid  -> contiguous
            out[(size_t)t * BATCH * 4 + (size_t)sBase * 4 + tid] = hxv;
        }
    }

    float* hxo = out + (size_t)T_STEPS * BATCH * 4;
    float* cxo = hxo + (size_t)BATCH * 4;
    hxo[(size_t)sBase * 4 + tid] = hxv;
    cxo[(size_t)sBase * 4 + tid] = cxv;
}

extern "C" void kernel_launch(void* const* d_in, const int* in_sizes, int n_in,
                              void* d_out, int out_size, void* d_ws, size_t ws_size,
                              hipStream_t stream) {
    (void)in_sizes; (void)n_in; (void)out_size; (void)ws_size;
    const float* x    = (const float*)d_in[0];   // [512,4096,16]
    const float* Wm   = (const float*)d_in[1];   // [4,4,20]
    const float* bias = (const float*)d_in[2];   // [4,4]
    const float* qrx  = (const float*)d_in[3];   // [4,4]
    const float* qry  = (const float*)d_in[4];
    const float* qrz  = (const float*)d_in[5];
    float* U = (float*)d_ws;                     // 4*8*16*2 floats = 4 KB scratch

    qlstm_setup_kernel<<<1, 32, 0, stream>>>(qrx, qry, qrz, U);
    qlstm_main_kernel<<<BATCH / 32, 128, 0, stream>>>(x, Wm, bias, U, (float*)d_out);
}